// Integrator_layer_80539226734962
// MI455X (gfx1250) — compile-verified
//
#include <hip/hip_runtime.h>

#ifndef __has_builtin
#define __has_builtin(x) 0
#endif

#if __has_builtin(__builtin_amdgcn_global_load_async_to_lds_b64)
#define USE_ASYNC 1
#else
#define USE_ASYNC 0
#endif

#if USE_ASYNC
#if __has_builtin(__builtin_amdgcn_s_wait_asynccnt)
#define WAIT_ASYNC(n) __builtin_amdgcn_s_wait_asynccnt(n)
#else
#define WAIT_ASYNC(n) asm volatile("s_wait_asynccnt %0" ::"n"(n) : "memory")
#endif
// The builtin's params are v2i pointers in addrspace(1)/(3). Cast via integer
// round-trip; low 32 bits of a generic LDS pointer are the LDS byte offset.
typedef int v2i_t __attribute__((vector_size(8)));
#define GLB1(p) ((__attribute__((address_space(1))) v2i_t*)(unsigned long long)(p))
#define LDS3(p) ((__attribute__((address_space(3))) v2i_t*)(unsigned)(unsigned long long)(p))
#define NBUF 8   // ring of LDS tile buffers (power of two)
#define DEPTH 6  // stages in flight; NBUF >= DEPTH+2 allows single barrier/iter
#else
#define NBUF 1
#endif

// Problem constants: inputs (4, 200, 64, 64, 8) f32, WINDOW=100, VM_THRESHOLD=2.0
// in flat:  (b*200+t)*32768 + h*512 + (w*8+c)
// out flat: (b*200+t)*32768 + (w*8+c)*64 + h        (out = swapaxes(spikes,1,-1))
// Per block: one b, one 100-step window, 32 h-rows, 32 wc-columns. 256 threads,
// 4 neurons/thread. Sequential IAF scan over t within the window.

#define PLANE 32768
#define LROW 34  // LDS row stride in floats: 8B-aligned rows, bank-conflict-free

__global__ __launch_bounds__(256) void iaf_spike_kernel(const float* __restrict__ in,
                                                        float* __restrict__ out)
{
    __shared__ float lbuf[NBUF][32 * LROW];

    const int tid = threadIdx.x;
    const int bid = blockIdx.x;
    const int wct = bid & 15;         // 16 wc tiles of 32
    const int ht  = (bid >> 4) & 1;   // 2 h tiles of 32
    const int win = (bid >> 5) & 1;   // 2 time windows of 100
    const int b   = bid >> 6;         // 4 batches

    const int wc0  = wct << 5;
    const int h0   = ht << 5;
    const int t0   = win * 100;
    const int tend = t0 + 100;

    // ---- staging layout: thread copies two 8B chunks (rows r0 and r0+16) ----
    const int r0 = tid >> 4;          // h-row within tile (0..15)
    const int o  = tid & 15;          // 8B chunk within the 128B row

    const float* gin = in + (size_t)b * 200 * PLANE
                          + (size_t)h0 * 512 + (size_t)wc0 + (size_t)o * 2;

    // ---- compute/store layout: lane = h, grp selects wc (grp + 8k, k=0..3) ----
    const int lane = tid & 31;
    const int grp  = tid >> 5;
    float* gout = out + (size_t)b * 200 * PLANE
                      + (size_t)(wc0 + grp) * 64 + (size_t)h0 + (size_t)lane;

    const int lidx = lane * LROW + grp;
    const int s0   = r0 * LROW + o * 2;
    const int s1   = s0 + 16 * LROW;

    float V0 = 0.f, V1 = 0.f, V2 = 0.f, V3 = 0.f;

#define COMPUTE_AND_STORE(cb, t)                                                 \
    do {                                                                         \
        float x0 = lbuf[cb][lidx + 0];                                           \
        float x1 = lbuf[cb][lidx + 8];                                           \
        float x2 = lbuf[cb][lidx + 16];                                          \
        float x3 = lbuf[cb][lidx + 24];                                          \
        V0 += x0; V1 += x1; V2 += x2; V3 += x3;                                  \
        float p0 = (V0 > 2.0f) ? 1.0f : 0.0f; V0 = (V0 > 2.0f) ? 0.0f : V0;      \
        float p1 = (V1 > 2.0f) ? 1.0f : 0.0f; V1 = (V1 > 2.0f) ? 0.0f : V1;      \
        float p2 = (V2 > 2.0f) ? 1.0f : 0.0f; V2 = (V2 > 2.0f) ? 0.0f : V2;      \
        float p3 = (V3 > 2.0f) ? 1.0f : 0.0f; V3 = (V3 > 2.0f) ? 0.0f : V3;      \
        float* op = gout + (size_t)(t) * PLANE;                                  \
        __builtin_nontemporal_store(p0, op);                                     \
        __builtin_nontemporal_store(p1, op + 512);                               \
        __builtin_nontemporal_store(p2, op + 1024);                              \
        __builtin_nontemporal_store(p3, op + 1536);                              \
    } while (0)

#if USE_ASYNC
#define STAGE(bi, t)                                                             \
    do {                                                                         \
        __builtin_amdgcn_global_load_async_to_lds_b64(                           \
            GLB1(gin + (size_t)(t) * PLANE + (size_t)r0 * 512),                  \
            LDS3(&lbuf[bi][s0]), 0, 0);                                          \
        __builtin_amdgcn_global_load_async_to_lds_b64(                           \
            GLB1(gin + (size_t)(t) * PLANE + (size_t)(r0 + 16) * 512),           \
            LDS3(&lbuf[bi][s1]), 0, 0);                                          \
    } while (0)

    // Prologue: stage DEPTH tiles ahead.
    for (int t = t0; t < t0 + DEPTH; ++t) STAGE((t - t0) & (NBUF - 1), t);

    // Single barrier per iteration: stage(t+DEPTH) targets the buffer read at
    // iteration t-(NBUF-DEPTH) (>= 2 back), so the barrier between the wait and
    // the compute already orders every read before the next overwrite.
    for (int t = t0; t < tend; ++t) {
        const int cb = (t - t0) & (NBUF - 1);
        if (t + DEPTH < tend) {
            STAGE((t - t0 + DEPTH) & (NBUF - 1), t + DEPTH);
            WAIT_ASYNC(2 * DEPTH);  // stage(t) complete (2 instrs/stage, in-order)
        } else {
            WAIT_ASYNC(0);          // drain at the tail
        }
        __syncthreads();            // publish stage(t) to all waves
        COMPUTE_AND_STORE(cb, t);
    }
#else
    // Fallback: register double-buffered global loads + LDS transpose.
    float2 c0 = {}, c1 = {}, n0 = {}, n1 = {};
    {
        const float* g = gin + (size_t)t0 * PLANE;
        c0 = *(const float2*)(g + (size_t)r0 * 512);
        c1 = *(const float2*)(g + (size_t)(r0 + 16) * 512);
    }
    for (int t = t0; t < tend; ++t) {
        __syncthreads();  // prior-iteration readers done before overwrite
        *(float2*)&lbuf[0][s0] = c0;
        *(float2*)&lbuf[0][s1] = c1;
        if (t + 1 < tend) {
            const float* g = gin + (size_t)(t + 1) * PLANE;
            n0 = *(const float2*)(g + (size_t)r0 * 512);
            n1 = *(const float2*)(g + (size_t)(r0 + 16) * 512);
        }
        __syncthreads();
        COMPUTE_AND_STORE(0, t);
        c0 = n0; c1 = n1;
    }
#endif
}

extern "C" void kernel_launch(void* const* d_in, const int* in_sizes, int n_in,
                              void* d_out, int out_size, void* d_ws, size_t ws_size,
                              hipStream_t stream) {
    (void)in_sizes; (void)n_in; (void)out_size; (void)d_ws; (void)ws_size;
    const float* in = (const float*)d_in[0];
    float* out = (float*)d_out;
    // 256 blocks: 4 (batch) x 2 (window) x 2 (h tile) x 16 (wc tile); 256 threads.
    iaf_spike_kernel<<<dim3(256), dim3(256), 0, stream>>>(in, out);
}